// GlueEdgeDGCNN_36541581754797
// MI455X (gfx1250) — compile-verified
//
#include <hip/hip_runtime.h>
#include <math.h>

// ---------------------------------------------------------------------------
// GlueEdgeDGCNN for MI455X (gfx1250): fp32 WMMA 16x16x4 GEMMs + kNN gather.
// Edge-conv factored: e@W = hi@(Wt-Wb) + hj@Wb  ->  two N x 128 x 128 GEMMs
// per layer + gather-max, instead of 1M x (256x128) edge GEMMs.
// ---------------------------------------------------------------------------

typedef float v2f __attribute__((ext_vector_type(2)));
typedef float v8f __attribute__((ext_vector_type(8)));

#define HID    128
#define KNN    16
#define NGRAPH 64
#define NNODE  1024
#define NTOT   (NGRAPH * NNODE)
#define INDIM  17
#define TEMB   32          // HID/4
#define ZDIM   48          // 16 feats + 32 temb

static __device__ __forceinline__ v8f wmma_k4(v2f a, v2f b, v8f c) {
  // V_WMMA_F32_16X16X4_F32: D = A(16x4) x B(4x16) + C(16x16)
  return __builtin_amdgcn_wmma_f32_16x16x4_f32(
      /*neg_a=*/false, a, /*neg_b=*/false, b,
      /*c_mod=*/(short)0, c, /*reuse_a=*/false, /*reuse_b=*/false);
}

static __device__ __forceinline__ float sigmoidf_(float x) {
  return 1.0f / (1.0f + __expf(-x));
}

// z[m][k]: k<16 -> feats = x[node, 1+k]; k>=16 -> t*te_w[k-16] + te_b[k-16]
static __device__ __forceinline__ float z_val(const float* __restrict__ x,
                                              int node, int k,
                                              const float* __restrict__ te_w,
                                              const float* __restrict__ te_b) {
  if (k < 16) return x[node * INDIM + 1 + k];
  int j = k - 16;
  return x[node * INDIM] * te_w[j] + te_b[j];
}

// ---------------------------------------------------------------------------
// Kernel 1: h = GLU([feats, t_emb]) : (N x 48) @ (48 x 128), two gates.
// Block = 256 (8 waves). Block -> 16-node M tile, wave -> 16-col N tile.
// ---------------------------------------------------------------------------
__global__ __launch_bounds__(256) void k_glu_embed(
    const float* __restrict__ x,
    const float* __restrict__ te_w, const float* __restrict__ te_b,
    const float* __restrict__ lw, const float* __restrict__ lb,
    const float* __restrict__ gw, const float* __restrict__ gb,
    float* __restrict__ h) {
  const int tid  = threadIdx.x;
  const int lane = tid & 31;
  const int wave = tid >> 5;          // N tile: 0..7
  const int half = lane >> 4;         // 0 or 1 (K sub-pair)
  const int ln   = lane & 15;
  const int m0   = blockIdx.x * 16;   // first node of the tile
  const int ncol = wave * 16 + ln;    // output column 0..127
  const int row  = m0 + ln;           // A row for this lane

  v8f acc_l = {};
  v8f acc_g = {};
  #pragma unroll
  for (int c = 0; c < ZDIM / 4; ++c) {       // 12 K-chunks of 4
    const int k0 = c * 4 + half * 2;
    v2f av; av.x = z_val(x, row, k0, te_w, te_b);
            av.y = z_val(x, row, k0 + 1, te_w, te_b);
    v2f bl; bl.x = lw[k0 * HID + ncol]; bl.y = lw[(k0 + 1) * HID + ncol];
    v2f bg; bg.x = gw[k0 * HID + ncol]; bg.y = gw[(k0 + 1) * HID + ncol];
    acc_l = wmma_k4(av, bl, acc_l);
    acc_g = wmma_k4(av, bg, acc_g);
  }

  const float bias_l = lb[ncol];
  const float bias_g = gb[ncol];
  #pragma unroll
  for (int r = 0; r < 8; ++r) {              // C/D: row = r + 8*half
    const float lv = acc_l[r] + bias_l;
    const float gv = acc_g[r] + bias_g;
    const int out_row = m0 + r + half * 8;
    h[out_row * HID + ncol] = lv * sigmoidf_(gv);
  }
}

// ---------------------------------------------------------------------------
// Kernel 2: per-node k=16 nearest neighbors by |t_i - t_j| within a graph.
// Graph t values staged in LDS; best-16 list kept in registers (unrolled).
// Grid = 64 graphs * 4 sub-blocks, block = 256 threads (1 node/thread).
// ---------------------------------------------------------------------------
__global__ __launch_bounds__(256) void k_topk(const float* __restrict__ x,
                                              int* __restrict__ idx) {
  __shared__ float tsh[NNODE];
  const int g   = blockIdx.x >> 2;
  const int sub = blockIdx.x & 3;
  const int tid = threadIdx.x;

  for (int j = tid; j < NNODE; j += 256)
    tsh[j] = x[(g * NNODE + j) * INDIM];     // key_feat = x[:,0]
  __syncthreads();

  const int il = sub * 256 + tid;            // graph-local node id
  const float ti = tsh[il];

  float bd[KNN];
  int   bi[KNN];
  #pragma unroll
  for (int s = 0; s < KNN; ++s) { bd[s] = 3.0e38f; bi[s] = 0; }

  for (int j = 0; j < NNODE; ++j) {
    if (j == il) continue;                   // diagonal excluded (BIG in ref)
    const float d = fabsf(ti - tsh[j]);
    if (d < bd[KNN - 1]) {
      bd[KNN - 1] = d; bi[KNN - 1] = j;
      #pragma unroll
      for (int s = KNN - 1; s > 0; --s) {    // one bubble pass re-sorts
        if (bd[s] < bd[s - 1]) {
          float td = bd[s]; bd[s] = bd[s - 1]; bd[s - 1] = td;
          int   tj = bi[s]; bi[s] = bi[s - 1]; bi[s - 1] = tj;
        }
      }
    }
  }

  const int node = g * NNODE + il;
  #pragma unroll
  for (int s = 0; s < KNN; ++s) idx[node * KNN + s] = g * NNODE + bi[s];
}

// ---------------------------------------------------------------------------
// Kernel 3/5: dual GEMM for an edge-conv layer.
//   a  = in @ (W_top - W_bot) + b     (16x16x4 WMMA, K=128)
//   bj = in @ W_bot
// W is (256,128) row-major: rows 0..127 = W_top, 128..255 = W_bot.
// ---------------------------------------------------------------------------
__global__ __launch_bounds__(256) void k_edge_gemm(
    const float* __restrict__ in, const float* __restrict__ w,
    const float* __restrict__ b,
    float* __restrict__ outa, float* __restrict__ outbj) {
  const int tid  = threadIdx.x;
  const int lane = tid & 31;
  const int wave = tid >> 5;
  const int half = lane >> 4;
  const int ln   = lane & 15;
  const int m0   = blockIdx.x * 16;
  const int ncol = wave * 16 + ln;
  const int row  = m0 + ln;

  v8f acc_a = {};
  v8f acc_j = {};
  #pragma unroll
  for (int c = 0; c < HID / 4; ++c) {        // 32 K-chunks
    const int k0 = c * 4 + half * 2;
    v2f av; av.x = in[row * HID + k0]; av.y = in[row * HID + k0 + 1];
    const float wt0 = w[k0 * HID + ncol];
    const float wt1 = w[(k0 + 1) * HID + ncol];
    const float wb0 = w[(k0 + 128) * HID + ncol];
    const float wb1 = w[(k0 + 129) * HID + ncol];
    v2f bt; bt.x = wt0 - wb0; bt.y = wt1 - wb1;  // hi coefficient
    v2f bb; bb.x = wb0;       bb.y = wb1;        // hj coefficient
    acc_a = wmma_k4(av, bt, acc_a);
    acc_j = wmma_k4(av, bb, acc_j);
  }

  const float bias = b[ncol];
  #pragma unroll
  for (int r = 0; r < 8; ++r) {
    const int out_row = m0 + r + half * 8;
    outa[out_row * HID + ncol]  = acc_a[r] + bias;
    outbj[out_row * HID + ncol] = acc_j[r];
  }
}

// ---------------------------------------------------------------------------
// Kernel 4/6: out[i] = relu(a[i] + max_k bj[idx[i,k]])  (float4 per lane).
// Block = 256 = 8 nodes x 32 lanes; lane covers 4 channels.
// ---------------------------------------------------------------------------
__global__ __launch_bounds__(256) void k_gather_max(
    const float* __restrict__ a, const float* __restrict__ bj,
    const int* __restrict__ idx, float* __restrict__ out) {
  const int tid  = threadIdx.x;
  const int node = blockIdx.x * 8 + (tid >> 5);
  const int lane = tid & 31;
  const int c0   = lane * 4;

  float4 m = make_float4(-3.0e38f, -3.0e38f, -3.0e38f, -3.0e38f);
  const int* ip = idx + node * KNN;
  #pragma unroll
  for (int k = 0; k < KNN; ++k) {
    const int j = ip[k];
    const float4 v = *reinterpret_cast<const float4*>(bj + (size_t)j * HID + c0);
    m.x = fmaxf(m.x, v.x); m.y = fmaxf(m.y, v.y);
    m.z = fmaxf(m.z, v.z); m.w = fmaxf(m.w, v.w);
  }
  const float4 av = *reinterpret_cast<const float4*>(a + (size_t)node * HID + c0);
  float4 o;
  o.x = fmaxf(av.x + m.x, 0.0f);
  o.y = fmaxf(av.y + m.y, 0.0f);
  o.z = fmaxf(av.z + m.z, 0.0f);
  o.w = fmaxf(av.w + m.w, 0.0f);
  *reinterpret_cast<float4*>(out + (size_t)node * HID + c0) = o;
}

// ---------------------------------------------------------------------------
// Kernel 7: graph pooling. comb = [x1, x2] (1024 x 256) -> [max(256), mean(256)].
// ---------------------------------------------------------------------------
__global__ __launch_bounds__(256) void k_pool(const float* __restrict__ x1,
                                              const float* __restrict__ x2,
                                              float* __restrict__ pooled) {
  const int g = blockIdx.x;
  const int t = threadIdx.x;                 // channel of comb, 0..255
  const float* src = (t < HID) ? x1 : x2;
  const int ch = t & (HID - 1);
  float mx = -3.0e38f, sm = 0.0f;
  for (int i = 0; i < NNODE; ++i) {
    const float v = src[(size_t)(g * NNODE + i) * HID + ch];
    mx = fmaxf(mx, v);
    sm += v;
  }
  pooled[g * 512 + t]       = mx;
  pooled[g * 512 + 256 + t] = sm * (1.0f / NNODE);
}

// ---------------------------------------------------------------------------
// Kernel 8: head. logits = GLU(pooled) @ out_w + out_b ; log_softmax over 2.
// ---------------------------------------------------------------------------
__global__ __launch_bounds__(128) void k_head(
    const float* __restrict__ pooled,
    const float* __restrict__ lw, const float* __restrict__ lb,
    const float* __restrict__ gw, const float* __restrict__ gb,
    const float* __restrict__ ow, const float* __restrict__ ob,
    float* __restrict__ out) {
  __shared__ float hsh[HID];
  __shared__ float lsh[2];
  const int g = blockIdx.x;
  const int c = threadIdx.x;

  float lin = lb[c], gate = gb[c];
  for (int j = 0; j < 512; ++j) {
    const float p = pooled[g * 512 + j];
    lin  += p * lw[j * HID + c];
    gate += p * gw[j * HID + c];
  }
  hsh[c] = lin * sigmoidf_(gate);
  __syncthreads();

  if (c < 2) {
    float acc = ob[c];
    for (int j = 0; j < HID; ++j) acc += hsh[j] * ow[j * 2 + c];
    lsh[c] = acc;
  }
  __syncthreads();

  if (c == 0) {
    const float l0 = lsh[0], l1 = lsh[1];
    const float mx = fmaxf(l0, l1);
    const float lse = mx + __logf(__expf(l0 - mx) + __expf(l1 - mx));
    out[g * 2 + 0] = l0 - lse;
    out[g * 2 + 1] = l1 - lse;
  }
}

// ---------------------------------------------------------------------------
// Launch
// ---------------------------------------------------------------------------
extern "C" void kernel_launch(void* const* d_in, const int* in_sizes, int n_in,
                              void* d_out, int out_size, void* d_ws, size_t ws_size,
                              hipStream_t stream) {
  const float* x     = (const float*)d_in[0];
  const float* te_w  = (const float*)d_in[1];
  const float* te_b  = (const float*)d_in[2];
  const float* ig_lw = (const float*)d_in[3];
  const float* ig_lb = (const float*)d_in[4];
  const float* ig_gw = (const float*)d_in[5];
  const float* ig_gb = (const float*)d_in[6];
  const float* c1_w  = (const float*)d_in[7];
  const float* c1_b  = (const float*)d_in[8];
  const float* c2_w  = (const float*)d_in[9];
  const float* c2_b  = (const float*)d_in[10];
  const float* fg_lw = (const float*)d_in[11];
  const float* fg_lb = (const float*)d_in[12];
  const float* fg_gw = (const float*)d_in[13];
  const float* fg_gb = (const float*)d_in[14];
  const float* out_w = (const float*)d_in[15];
  const float* out_b = (const float*)d_in[16];
  // d_in[17] = batch (unused; nodes are contiguous per graph)
  // d_in[18] = num_graphs (64, compile-time constant here)

  // Workspace layout (bytes). h's buffer is dead after conv1 GEMM -> reused
  // for x2.
  const size_t SZ_MAT = (size_t)NTOT * HID * sizeof(float);   // 32 MB
  char* wsb = (char*)d_ws;
  float* h      = (float*)(wsb);
  int*   idx    = (int*)  (wsb + SZ_MAT);
  float* a_buf  = (float*)(wsb + SZ_MAT + (size_t)NTOT * KNN * sizeof(int));
  float* bj_buf = (float*)((char*)a_buf + SZ_MAT);
  float* x1     = (float*)((char*)bj_buf + SZ_MAT);
  float* pooled = (float*)((char*)x1 + SZ_MAT);
  float* x2     = h;                                          // reuse

  float* out = (float*)d_out;

  // 1) h = GLU([feats, t_emb])
  k_glu_embed<<<NTOT / 16, 256, 0, stream>>>(x, te_w, te_b, ig_lw, ig_lb,
                                             ig_gw, ig_gb, h);
  // 2) kNN indices on |t_i - t_j| (shared by both conv layers)
  k_topk<<<NGRAPH * 4, 256, 0, stream>>>(x, idx);

  // 3) conv1: a = h@(Wt-Wb)+b, bj = h@Wb ; x1 = relu(a + max_k bj[idx])
  k_edge_gemm<<<NTOT / 16, 256, 0, stream>>>(h, c1_w, c1_b, a_buf, bj_buf);
  k_gather_max<<<NTOT / 8, 256, 0, stream>>>(a_buf, bj_buf, idx, x1);

  // 4) conv2 on x1 -> x2
  k_edge_gemm<<<NTOT / 16, 256, 0, stream>>>(x1, c2_w, c2_b, a_buf, bj_buf);
  k_gather_max<<<NTOT / 8, 256, 0, stream>>>(a_buf, bj_buf, idx, x2);

  // 5) pooling + head
  k_pool<<<NGRAPH, 256, 0, stream>>>(x1, x2, pooled);
  k_head<<<NGRAPH, 128, 0, stream>>>(pooled, fg_lw, fg_lb, fg_gw, fg_gb,
                                     out_w, out_b, out);
}